// TensorProduct_54168127537278
// MI455X (gfx1250) — compile-verified
//
#include <hip/hip_runtime.h>

// N = 32768 edges, MUL = 512 channels; x1 row = 2048 floats, out row = 2048 floats.
typedef __attribute__((ext_vector_type(2))) float v2f;
typedef __attribute__((ext_vector_type(4))) float v4f;
typedef __attribute__((ext_vector_type(8))) float v8f;

// gcc-style int4 to exactly match the async-LDS builtin's pointee type
typedef int v4i __attribute__((vector_size(16)));
typedef __attribute__((address_space(1))) v4i as1_v4i;   // printed as "__device__" by hipcc
typedef __attribute__((address_space(3))) v4i as3_v4i;   // LDS

#if __has_builtin(__builtin_amdgcn_global_load_async_to_lds_b128)
#define HAVE_ASYNC_LDS 1
#else
#define HAVE_ASYNC_LDS 0
#endif

// ---------------------------------------------------------------------------
// Kernel 1: reduce the 5 weight blocks (512 each) to 5 scalars with the CG
// normalizations folded in:  c = {w0, w1/sqrt(3), w2, w3, w4/sqrt(2)}
// ---------------------------------------------------------------------------
__global__ __launch_bounds__(160) void tp_wsum_kernel(const float* __restrict__ w,
                                                      float* __restrict__ c5) {
  const int wave = threadIdx.x >> 5;   // 0..4, one wave per instruction
  const int lane = threadIdx.x & 31;
  float s = 0.f;
  const float* wp = w + wave * 512;
#pragma unroll
  for (int i = 0; i < 16; ++i) s += wp[lane + 32 * i];
#pragma unroll
  for (int off = 16; off > 0; off >>= 1) s += __shfl_xor(s, off, 32);
  if (lane == 0) {
    float nm = 1.f;
    if (wave == 1) nm = 0.57735026918962576f;   // 1/sqrt(3)
    if (wave == 4) nm = 0.70710678118654752f;   // 1/sqrt(2)
    c5[wave] = s * nm;
  }
}

// ---------------------------------------------------------------------------
// Kernel 2: one block per edge. out_row = a(512x4) @ M(4x4) done as
// D(16x16) = A(16x4 := M^T in rows 0-3 AND 8-11) x B(4x16 := comps x channels)
// via V_WMMA_F32_16X16X4_F32, 8 tiles per wave, LDS updated in place.
// ---------------------------------------------------------------------------
__global__ __launch_bounds__(128) void tp_main_kernel(const float* __restrict__ x1,
                                                      const float* __restrict__ x2,
                                                      const float* __restrict__ c5,
                                                      float* __restrict__ out) {
  __shared__ __align__(16) float lin[2048];   // x1 row, rewritten in place with out row
  __shared__ float Ml[16];                    // M[k][m] at Ml[k*4+m]

  const int n    = blockIdx.x;
  const int tid  = threadIdx.x;
  const int lane = tid & 31;
  const int wave = tid >> 5;

  // ---- stage x1 row (8 KB) into LDS
#if HAVE_ASYNC_LDS
  {
    as1_v4i* gsrc = (as1_v4i*)(x1 + (size_t)n * 2048);
    as3_v4i* ldst = (as3_v4i*)lin;
#pragma unroll
    for (int i = 0; i < 4; ++i) {
      const int e = tid + 128 * i;           // 16-byte elements
      __builtin_amdgcn_global_load_async_to_lds_b128(gsrc + e, ldst + e, 0, 0);
    }
  }
#else
  {
    const v4f* src = (const v4f*)(x1 + (size_t)n * 2048);
    v4f* dst = (v4f*)lin;
#pragma unroll
    for (int i = 0; i < 4; ++i)
      dst[tid + 128 * i] = __builtin_nontemporal_load(src + tid + 128 * i);
  }
#endif

  // ---- build per-edge 4x4 mixing matrix M[k][m] (k = input comp, m = out comp)
  if (tid < 16) {
    const float c0 = c5[0], c1 = c5[1], c2 = c5[2], c3 = c5[3], c4 = c5[4];
    const float q0 = x2[n * 4 + 0];   // s2
    const float q1 = x2[n * 4 + 1];   // v2x
    const float q2 = x2[n * 4 + 2];   // v2y
    const float q3 = x2[n * 4 + 3];   // v2z
    const int k = tid >> 2, m = tid & 3;
    float val;
    if (k == 0) {                     // s1 row: c0*s2 into scalar, c2*v2 into vec
      val = (m == 0) ? c0 * q0 : c2 * ((m == 1) ? q1 : (m == 2) ? q2 : q3);
    } else if (m == 0) {              // v1 -> scalar: c1 * v2_k
      val = c1 * ((k == 1) ? q1 : (k == 2) ? q2 : q3);
    } else if (k == m) {              // v1 -> vec diagonal: c3 * s2
      val = c3 * q0;
    } else {                          // cross product: sign * c4 * v2_jj
      const int kk = k - 1, mm = m - 1;
      const int jj = 3 - kk - mm;
      const float sgn = (((jj - kk + 3) % 3) == 1) ? 1.f : -1.f;
      const float qj = (jj == 0) ? q1 : (jj == 1) ? q2 : q3;
      val = sgn * c4 * qj;
    }
    Ml[k * 4 + m] = val;
  }

#if HAVE_ASYNC_LDS
#if __has_builtin(__builtin_amdgcn_s_wait_asynccnt)
  __builtin_amdgcn_s_wait_asynccnt(0);
#else
  asm volatile("s_wait_asynccnt 0" ::: "memory");
#endif
#endif
  __syncthreads();

  // ---- A fragment: lanes 0-15 hold rows M=lane (K=0,1), lanes 16-31 rows
  //      M=lane-16 (K=2,3). M^T occupies rows 0-3 AND (duplicated) rows 8-11,
  //      so D rows r and r+8 are equal -> all 32 lanes carry valid outputs.
  const int  l16 = lane & 15;
  const int  hi  = lane >> 4;              // 0 = lanes 0-15, 1 = lanes 16-31
  const bool av  = (l16 & 4) == 0;         // rows 0-3 and 8-11 are nonzero
  const int  ab  = (hi << 3) + (l16 & 3);  // Ml[k*4+m]: k base 0 (lo) / 2 (hi)
  v2f A;
  A[0] = av ? Ml[ab]     : 0.f;            // K = 0 (lo) / K = 2 (hi)
  A[1] = av ? Ml[ab + 4] : 0.f;            // K = 1 (lo) / K = 3 (hi)

  // ---- preload all 8 B fragments (branchless addresses), then 8 WMMAs with
  //      in-place stores: load addresses == store addresses by construction.
  //      lo lanes: (s1[u], v1x[u])  ->  (o_s[u], o_x[u])
  //      hi lanes: (v1y[u], v1z[u]) ->  (o_y[u], o_z[u])
  int adr0[8], adr1[8];
  v2f Bf[8];
#pragma unroll
  for (int t = 0; t < 8; ++t) {
    const int u = (wave * 8 + t) * 16 + l16;
    adr0[t] = u + hi * (513 + 2 * u);      // lo: u          hi: 512+3u+1
    adr1[t] = 512 + 3 * u + 2 * hi;        // lo: 512+3u     hi: 512+3u+2
    Bf[t][0] = lin[adr0[t]];
    Bf[t][1] = lin[adr1[t]];
  }
#pragma unroll
  for (int t = 0; t < 8; ++t) {
    v8f C = {};
    v8f D = __builtin_amdgcn_wmma_f32_16x16x4_f32(false, A, false, Bf[t],
                                                  (short)0, C, false, false);
    lin[adr0[t]] = hi ? D[2] : D[0];       // o_s (lo) / o_y (hi)
    lin[adr1[t]] = hi ? D[3] : D[1];       // o_x (lo) / o_z (hi)
  }
  __syncthreads();

  // ---- coalesced 128-bit non-temporal stores of the finished row
  {
    const v4f* src = (const v4f*)lin;
    v4f* dst = (v4f*)(out + (size_t)n * 2048);
#pragma unroll
    for (int i = 0; i < 4; ++i)
      __builtin_nontemporal_store(src[tid + 128 * i], dst + tid + 128 * i);
  }
}

// ---------------------------------------------------------------------------
extern "C" void kernel_launch(void* const* d_in, const int* in_sizes, int n_in,
                              void* d_out, int out_size, void* d_ws, size_t ws_size,
                              hipStream_t stream) {
  const float* x1 = (const float*)d_in[0];   // (N, 2048) f32
  const float* x2 = (const float*)d_in[1];   // (N, 4)    f32
  const float* w  = (const float*)d_in[2];   // (2560,)   f32
  float* out = (float*)d_out;                // (N, 2048) f32
  float* c5  = (float*)d_ws;                 // 5 reduced weight scalars

  const int n = in_sizes[0] / 2048;          // N = 32768

  tp_wsum_kernel<<<1, 160, 0, stream>>>(w, c5);
  tp_main_kernel<<<n, 128, 0, stream>>>(x1, x2, c5, out);
}